// AttentionWeightedCELoss_68616397521419
// MI455X (gfx1250) — compile-verified
//
#include <hip/hip_runtime.h>

// AttentionWeightedCELoss for MI455X (gfx1250, wave32).
// Single pass over logits: per-pixel softmax stats -> per-class register
// accumulators -> wave32 shfl reduction -> LDS -> replicated global atomics
// (16 replicas to kill L2 atomic serialization) -> tiny finalize wave.
// global_prefetch_b8 warms L2 one grid-stride ahead of the 10 channel streams.

#define NUM_C   10
#define HW_     (512 * 512)
#define NPIX    (8 * HW_)
#define NGROUP  (NPIX / 4)
#define HW4     (HW_ / 4)
#define NREP    16            // replicated accumulator banks in d_ws
#define REPSTR  32            // floats per replica bank (30 used, padded)
#define GRID    1024
#define BLOCK   256

__global__ void awce_init(float* __restrict__ ws) {
    const int i = threadIdx.x;   // one block of NREP*REPSTR threads
    if (i < NREP * REPSTR) ws[i] = 0.0f;
}

__global__ __launch_bounds__(BLOCK) void awce_main(const float* __restrict__ logits,
                                                   const int*   __restrict__ targets,
                                                   float*       __restrict__ ws) {
    const float4* __restrict__ lg4 = reinterpret_cast<const float4*>(logits);
    const int4*   __restrict__ tg4 = reinterpret_cast<const int4*>(targets);

    float cnt[NUM_C], entS[NUM_C], ceS[NUM_C];
#pragma unroll
    for (int c = 0; c < NUM_C; ++c) { cnt[c] = 0.f; entS[c] = 0.f; ceS[c] = 0.f; }

    const int tid    = blockIdx.x * blockDim.x + threadIdx.x;
    const int stride = gridDim.x * blockDim.x;

    for (int g = tid; g < NGROUP; g += stride) {
        const int p     = g << 2;                 // pixel index (quad base)
        const int b     = p >> 18;                // p / HW_
        const int hw4   = (p & (HW_ - 1)) >> 2;   // (p % HW_) / 4
        const int base4 = b * NUM_C * HW4 + hw4;

        // Prefetch next grid-stride tile (gfx1250 global_prefetch_b8):
        // warms L2 (192 MB; whole tensor fits) ~42 MB ahead of use.
        const int gn = g + stride;
        if (gn < NGROUP) {
            const int pn     = gn << 2;
            const int bn     = pn >> 18;
            const int hwn4   = (pn & (HW_ - 1)) >> 2;
            const int basen4 = bn * NUM_C * HW4 + hwn4;
#pragma unroll
            for (int c = 0; c < NUM_C; ++c)
                __builtin_prefetch((const void*)(lg4 + basen4 + c * HW4), 0, 1);
        }

        // 10 channels x 4 pixels via global_load_b128 (coalesced 512B/wave/channel).
        float l[4][NUM_C];
#pragma unroll
        for (int c = 0; c < NUM_C; ++c) {
            const float4 q = lg4[base4 + c * HW4];
            l[0][c] = q.x; l[1][c] = q.y; l[2][c] = q.z; l[3][c] = q.w;
        }
        const int4 t4 = tg4[g];
        const int ts[4] = { t4.x, t4.y, t4.z, t4.w };

#pragma unroll
        for (int k = 0; k < 4; ++k) {
            float m = l[k][0];
#pragma unroll
            for (int c = 1; c < NUM_C; ++c) m = fmaxf(m, l[k][c]);

            float s = 0.f, wsum = 0.f;
            float d[NUM_C];
#pragma unroll
            for (int c = 0; c < NUM_C; ++c) {
                d[c] = l[k][c] - m;
                const float e = __expf(d[c]);
                s    += e;
                wsum  = fmaf(e, d[c], wsum);       // sum e_c * (l_c - m)
            }
            const float logS    = __logf(s);
            const float entropy = logS - wsum / s; // -sum p*log p
            const int   t       = ts[k];

            // Branchless scatter into 30 statically-indexed register accumulators.
            // t == 10 (ignore_index) matches no class -> contributes nothing.
#pragma unroll
            for (int c = 0; c < NUM_C; ++c) {
                const float match = (t == c) ? 1.0f : 0.0f;
                cnt[c]  += match;
                entS[c]  = fmaf(match, entropy,     entS[c]);
                ceS[c]   = fmaf(match, logS - d[c], ceS[c]);  // ce = logS + m - l_t
            }
        }
    }

    // wave32 butterfly reduction of all 30 accumulators
#pragma unroll
    for (int i = 0; i < NUM_C; ++i) {
#pragma unroll
        for (int off = 16; off > 0; off >>= 1) {
            cnt[i]  += __shfl_xor(cnt[i],  off, 32);
            entS[i] += __shfl_xor(entS[i], off, 32);
            ceS[i]  += __shfl_xor(ceS[i],  off, 32);
        }
    }

    __shared__ float sm[3 * NUM_C][8];
    const int wave = threadIdx.x >> 5;
    const int lane = threadIdx.x & 31;
    if (lane == 0) {
#pragma unroll
        for (int i = 0; i < NUM_C; ++i) {
            sm[i][wave]             = cnt[i];
            sm[NUM_C + i][wave]     = entS[i];
            sm[2 * NUM_C + i][wave] = ceS[i];
        }
    }
    __syncthreads();

    if (threadIdx.x < 3 * NUM_C) {
        float s = 0.f;
#pragma unroll
        for (int w = 0; w < 8; ++w) s += sm[threadIdx.x][w];
        // Replicated banks: contention per address = GRID/NREP instead of GRID.
        atomicAdd(&ws[(blockIdx.x & (NREP - 1)) * REPSTR + threadIdx.x], s);
    }
}

__global__ void awce_final(const float* __restrict__ ws, float* __restrict__ out) {
    const int lane = threadIdx.x & 31;

    float cnt = 0.f, ent = 0.f, ces = 0.f;
    if (lane < NUM_C) {
#pragma unroll
        for (int r = 0; r < NREP; ++r) {
            cnt += ws[r * REPSTR + lane];
            ent += ws[r * REPSTR + NUM_C + lane];
            ces += ws[r * REPSTR + 2 * NUM_C + lane];
        }
    }

    float nv = cnt;
#pragma unroll
    for (int off = 16; off > 0; off >>= 1) nv += __shfl_xor(nv, off, 32);

    const float w_base = (cnt > 0.f && nv > 0.f) ? (nv - cnt) / fmaxf(nv, 1.f) : 0.f;
    const float em     = (cnt > 0.f) ? ent / fmaxf(cnt, 1.f) : 0.f;
    const float wc     = w_base * (1.f + 0.5f * em);

    float contrib = wc * ces;
#pragma unroll
    for (int off = 16; off > 0; off >>= 1) contrib += __shfl_xor(contrib, off, 32);

    if (threadIdx.x == 0) out[0] = contrib / (nv + 1e-6f);
}

extern "C" void kernel_launch(void* const* d_in, const int* in_sizes, int n_in,
                              void* d_out, int out_size, void* d_ws, size_t ws_size,
                              hipStream_t stream) {
    (void)in_sizes; (void)n_in; (void)out_size; (void)ws_size;
    const float* logits  = (const float*)d_in[0];
    const int*   targets = (const int*)d_in[1];
    float*       ws      = (float*)d_ws;
    float*       out     = (float*)d_out;

    awce_init <<<1, NREP * REPSTR, 0, stream>>>(ws);
    awce_main <<<GRID, BLOCK,      0, stream>>>(logits, targets, ws);
    awce_final<<<1, 32,            0, stream>>>(ws, out);
}